// RoIPooling2D_1606317769515
// MI455X (gfx1250) — compile-verified
//
#include <hip/hip_runtime.h>
#include <hip/hip_bf16.h>
#include <cmath>

#define ROI_OUTH 7
#define ROI_OUTW 7
#define ROI_SCALE (1.0f / 16.0f)
#define XC 128
#define XH 64
#define XW 64
#define BLOCK 64

typedef unsigned int u32;
typedef unsigned long long u64;
typedef u32 v4u __attribute__((ext_vector_type(4)));
typedef int  v4i __attribute__((ext_vector_type(4)));
typedef int  v8i __attribute__((ext_vector_type(8)));

#if defined(__has_builtin)
#if __has_builtin(__builtin_amdgcn_tensor_load_to_lds) && \
    __has_builtin(__builtin_amdgcn_s_wait_tensorcnt)
#define USE_TDM 1
#endif
#endif
#ifndef USE_TDM
#define USE_TDM 0
#endif

// One block = one (roi n, channel c). Stage the RoI bounding box for this
// channel into LDS with the CDNA5 Tensor Data Mover (one 2D-tile DMA,
// TENSORcnt) -- falling back to async global->LDS loads (ASYNCcnt) -- then
// 49 threads each max-reduce one output bin from LDS.
__global__ __launch_bounds__(BLOCK) void roi_pool_tdm_kernel(
    const float* __restrict__ x, const float* __restrict__ rois,
    float* __restrict__ out, int N)
{
    __shared__ float tile[XH * XW];   // 16 KB worst-case bbox (<< 320 KB/WGP)

    const int blk = blockIdx.x;
    const int c   = blk % XC;
    const int n   = blk / XC;
    const int tid = threadIdx.x;

    // ---- uniform RoI decode (rintf == round-to-nearest-even == jnp.round) ----
    const float* r = rois + (size_t)n * 5;
    const int b  = (int)r[0];
    const int xs = (int)rintf(r[1] * ROI_SCALE);
    const int ys = (int)rintf(r[2] * ROI_SCALE);
    const int xe = (int)rintf(r[3] * ROI_SCALE);
    const int ye = (int)rintf(r[4] * ROI_SCALE);

    const float roi_w = fmaxf((float)(xe - xs + 1), 1.0f);
    const float roi_h = fmaxf((float)(ye - ys + 1), 1.0f);
    const float bin_w = roi_w / (float)ROI_OUTW;
    const float bin_h = roi_h / (float)ROI_OUTH;

    // Clipped bounding box of all bins: rows [hs0,he0) x cols [ws0,we0).
    const int hs0 = min(max(ys, 0), XH);
    const int he0 = min(max(ye + 1, 0), XH);
    const int ws0 = min(max(xs, 0), XW);
    const int we0 = min(max(xe + 1, 0), XW);
    const int bh  = he0 - hs0;
    const int bw  = we0 - ws0;
    const bool nonempty = (bh > 0) && (bw > 0);

    const float* __restrict__ src =
        x + (((size_t)b * XC + c) * XH) * (size_t)XW;

    if (nonempty) {
#if USE_TDM
        if (tid < 32) {   // wave 0 issues the DMA once for the block
            const u64 ga = (u64)(const void*)(src + (size_t)hs0 * XW + ws0);
            // Low 32 bits of a generic LDS pointer are the LDS byte offset.
            const u32 lds_base = (u32)(u64)(const void*)&tile[0];

            // D# group 0: count=1 | lds_addr | global_addr[56:0] | type=2
            const v4u g0 = { 1u,
                             lds_base,
                             (u32)ga,
                             ((u32)(ga >> 32) & 0x01FFFFFFu) | (2u << 30) };
            // D# group 1: data_size=4B; tensor_dim0=bw, tensor_dim1=bh,
            // tile_dim0=bw, tile_dim1=bh, tensor_dim0_stride=XW; no pad /
            // iterate / multicast / atomic-barrier.
            const v8i g1 = { (int)(2u << 16),
                             (int)(((u32)bw & 0xFFFFu) << 16),
                             (int)(((u32)bh & 0xFFFFu) << 16),
                             (int)(((u32)bw & 0xFFFFu) << 16),
                             (int)((u32)bh & 0xFFFFu),
                             (int)XW,
                             0, 0 };
            const v4i z4 = { 0, 0, 0, 0 };   // groups 2/3: tile_dim3/4 = 0 (unused)
#if __clang_major__ >= 23
            const v8i z8 = { 0, 0, 0, 0, 0, 0, 0, 0 };
            __builtin_amdgcn_tensor_load_to_lds(g0, g1, z4, z4, z8, 0);
#else
            __builtin_amdgcn_tensor_load_to_lds(g0, g1, z4, z4, 0);
#endif
        }
        __builtin_amdgcn_s_wait_tensorcnt(0);
#else
        // Fallback: row-parallel async global->LDS loads (no integer division;
        // bw <= 64 == blockDim, lane == column within the row).
        for (int h = hs0; h < he0; ++h) {
            if (tid < bw) {
                const float* gp = src + h * XW + ws0 + tid;
                const unsigned lds_off =
                    (unsigned)(u64)(const void*)&tile[(h - hs0) * bw + tid];
                asm volatile("global_load_async_to_lds_b32 %0, %1, off"
                             :
                             : "v"(lds_off), "v"(gp)
                             : "memory");
            }
        }
        asm volatile("s_wait_asynccnt 0" ::: "memory");
#endif
    }
    __syncthreads();

    // ---- 49 bins, one per thread, max-reduce from LDS (compact stride bw) ----
    if (tid < ROI_OUTH * ROI_OUTW) {
        const int ph = tid / ROI_OUTW;
        const int pw = tid % ROI_OUTW;

        const int hs = min(max((int)floorf((float)ph * bin_h) + ys, 0), XH);
        const int he = min(max((int)ceilf((float)(ph + 1) * bin_h) + ys, 0), XH);
        const int ws = min(max((int)floorf((float)pw * bin_w) + xs, 0), XW);
        const int we = min(max((int)ceilf((float)(pw + 1) * bin_w) + xs, 0), XW);

        float m = -INFINITY;
        for (int h = hs; h < he; ++h) {
            const float* row = &tile[(h - hs0) * bw - ws0];
            for (int w = ws; w < we; ++w)
                m = fmaxf(m, row[w]);
        }
        // empty bin -> 0 (also maps NaN -> 0, matching jnp.where(isfinite,...,0))
        if (!(m > -INFINITY)) m = 0.0f;

        out[((size_t)((size_t)n * XC + c) * ROI_OUTH + ph) * ROI_OUTW + pw] = m;
    }
}

extern "C" void kernel_launch(void* const* d_in, const int* in_sizes, int n_in,
                              void* d_out, int out_size, void* d_ws, size_t ws_size,
                              hipStream_t stream) {
    const float* x    = (const float*)d_in[0];
    const float* rois = (const float*)d_in[1];
    float* out        = (float*)d_out;

    const int N = in_sizes[1] / 5;          // rois are [N,5]
    dim3 grid((unsigned)(N * XC));
    dim3 block(BLOCK);
    roi_pool_tdm_kernel<<<grid, block, 0, stream>>>(x, rois, out, N);
}